// StrategyEvolver_59931973648719
// MI455X (gfx1250) — compile-verified
//
#include <hip/hip_runtime.h>
#include <math.h>

// ---------------------------------------------------------------------------
// Fitness-ordered Gram-Schmidt re-orthogonalization, N=256 rows, D=2048 fp32.
// Single persistent workgroup: active 16-row block lives in LDS (320KB/WGP),
// inter-block projection runs on V_WMMA_F32_16X16X4_F32; LDS staging uses the
// gfx1250 async global<->LDS path (ASYNCcnt) when the toolchain exposes it.
// ---------------------------------------------------------------------------

#define N_ROWS   256
#define DIM      2048
#define BLK      16
#define NBLK     (N_ROWS / BLK)
#define TPB      256
#define NWAVES   (TPB / 32)
#define EPS      1e-8f
#define A_STRIDE 2052   // 2048 + 4 pad -> WMMA operand fetch hits 16 banks
#define F_STRIDE 132    // 128 + 4 pad
#define PANEL    128

typedef __attribute__((ext_vector_type(2))) float v2f;
typedef __attribute__((ext_vector_type(8))) float v8f;
typedef __attribute__((__vector_size__(4 * sizeof(int)))) int vi4;
typedef __attribute__((address_space(1))) vi4* gvi4p;   // global int4*
typedef __attribute__((address_space(3))) vi4* lvi4p;   // LDS int4* (32-bit)

#ifndef __has_builtin
#define __has_builtin(x) 0
#endif
#if __has_builtin(__builtin_amdgcn_global_load_async_to_lds_b128) && \
    __has_builtin(__builtin_amdgcn_global_store_async_from_lds_b128) && \
    __has_builtin(__builtin_amdgcn_s_wait_asynccnt)
#define USE_ASYNC_LDS 1
#else
#define USE_ASYNC_LDS 0
#endif

__device__ __forceinline__ void g2l_b128(const float* g, float* l) {
#if USE_ASYNC_LDS
  __builtin_amdgcn_global_load_async_to_lds_b128(
      (gvi4p)(uintptr_t)g, (lvi4p)(uint32_t)(uintptr_t)l, 0, 0);
#else
  *(float4*)l = *(const float4*)g;
#endif
}

__device__ __forceinline__ void l2g_b128(float* g, const float* l) {
#if USE_ASYNC_LDS
  __builtin_amdgcn_global_store_async_from_lds_b128(
      (gvi4p)(uintptr_t)g, (lvi4p)(uint32_t)(uintptr_t)l, 0, 0);
#else
  *(float4*)g = *(const float4*)l;
#endif
}

__device__ __forceinline__ void async_wait0() {
#if USE_ASYNC_LDS
  __builtin_amdgcn_s_wait_asynccnt(0);
#endif
}

// ws layout (floats): [0,256) row ||.||^2 by orig index
//                     [256,512) order (int bits), descending stable argsort
//                     [512,768) ordered clamped original norms

__global__ void row_normsq_kernel(const float* __restrict__ bank,
                                  float* __restrict__ ws) {
  __shared__ float red[TPB];
  const int row = blockIdx.x;
  const float* r = bank + (size_t)row * DIM;
  float acc = 0.f;
  for (int c = threadIdx.x; c < DIM; c += TPB) { float x = r[c]; acc += x * x; }
  red[threadIdx.x] = acc;
  __syncthreads();
  for (int s = TPB / 2; s > 0; s >>= 1) {
    if (threadIdx.x < s) red[threadIdx.x] += red[threadIdx.x + s];
    __syncthreads();
  }
  if (threadIdx.x == 0) ws[row] = red[0];
}

__global__ void sort_kernel(const float* __restrict__ fitness,
                            float* __restrict__ ws) {
  __shared__ float f[N_ROWS];
  const int i = threadIdx.x;
  f[i] = fitness[i];
  __syncthreads();
  const float fi = f[i];
  int rank = 0;
  for (int j = 0; j < N_ROWS; ++j) {
    const float fj = f[j];
    rank += (int)((fj > fi) || (fj == fi && j < i));  // stable descending
  }
  int* order = (int*)(ws + N_ROWS);
  order[rank] = i;
  __syncthreads();
  const int g = order[i];
  ws[2 * N_ROWS + i] = fmaxf(sqrtf(ws[g]), EPS);
}

__launch_bounds__(TPB, 1)
__global__ void gs_kernel(const float* __restrict__ bank,
                          const float* __restrict__ ws,
                          float* __restrict__ out) {
  __shared__ float lds_A[BLK][A_STRIDE];        // 128KB: active block rows
  __shared__ float lds_F[BLK][F_STRIDE];        // staged finalized-tile panel
  __shared__ float lds_Cp[NWAVES][BLK][BLK];    // per-wave partial dot tiles
  __shared__ float lds_coeff[BLK][BLK];         // negated projection coeffs
  __shared__ float lds_c1[BLK];                 // intra-block coeffs
  __shared__ float lds_red[TPB];
  __shared__ int   lds_order[N_ROWS];
  __shared__ float lds_norm[N_ROWS];
  __shared__ float lds_invsq[N_ROWS];

  const int tid   = threadIdx.x;
  const int wave  = tid >> 5;
  const int lane  = tid & 31;
  const int lhalf = lane >> 4;    // 0 | 1 (WMMA lane half)
  const int lrow  = lane & 15;
  const int fr    = tid >> 4;     // staging: this thread's tile row (0..15)
  const int fsub  = tid & 15;     // staging: 16 threads per row
  const int cg    = fsub * 8;     // staging: 8-float column group in a panel

  {
    const int* order = (const int*)(ws + N_ROWS);
    lds_order[tid] = order[tid];
    lds_norm[tid]  = ws[2 * N_ROWS + tid];
  }
  __syncthreads();

  for (int b = 0; b < NBLK; ++b) {
    const int base = b * BLK;

    // ---- Phase 0: load ordered block rows into LDS (b128, async path) ----
    {
      const float* asrc = bank + (size_t)lds_order[base + fr] * DIM;
#pragma unroll 4
      for (int j = 0; j < DIM / 4 / 16; ++j) {      // 32 x b128 per thread
        const int c4 = fsub + j * 16;               // interleaved -> coalesced
        g2l_b128(asrc + 4 * c4, &lds_A[fr][4 * c4]);
      }
      async_wait0();
    }
    __syncthreads();

    // ---- Phase 1: project out all previously finalized 16-row tiles ------
    for (int t = 0; t < b; ++t) {
      const int pbase = t * BLK;
      const float* fsrc = out + (size_t)lds_order[pbase + fr] * DIM;  // 1 lookup/tile

      // pass A: dot[16x16] = A_blk(16xD) @ F_tile(16xD)^T via WMMA f32 16x16x4
      v8f acc = {};
      for (int kp = 0; kp < DIM; kp += PANEL) {
        __syncthreads();
        g2l_b128(fsrc + kp + cg,     &lds_F[fr][cg]);
        g2l_b128(fsrc + kp + cg + 4, &lds_F[fr][cg + 4]);
        async_wait0();
        __syncthreads();
#pragma unroll
        for (int s = 0; s < 4; ++s) {
          const int kl = wave * 16 + s * 4;     // K offset inside panel
          const int k  = kp + kl;
          v2f a, bb;
          const float* ap = &lds_A[lrow][k + 2 * lhalf];
          a.x = ap[0]; a.y = ap[1];             // A[m][k..k+3] (ISA 16x4 layout)
          const float* bp = &lds_F[lrow][kl + 2 * lhalf];
          bb.x = bp[0]; bb.y = bp[1];           // B[k][n] = F[n][k]
          acc = __builtin_amdgcn_wmma_f32_16x16x4_f32(
              false, a, false, bb, (short)0, acc, false, false);
        }
      }
#pragma unroll
      for (int g = 0; g < 8; ++g) lds_Cp[wave][g + 8 * lhalf][lrow] = acc[g];
      __syncthreads();
      {
        const int m = tid >> 4, n = tid & 15;
        float s = 0.f;
#pragma unroll
        for (int w = 0; w < NWAVES; ++w) s += lds_Cp[w][m][n];
        lds_coeff[m][n] = -s * lds_invsq[pbase + n];   // negate: update is A += (-c)F
      }
      __syncthreads();

      // pass B: A_blk(16xD) += coeff(16x16) @ F_tile(16xD) via WMMA f32 16x16x4
      for (int kp = 0; kp < DIM; kp += PANEL) {
        __syncthreads();
        g2l_b128(fsrc + kp + cg,     &lds_F[fr][cg]);
        g2l_b128(fsrc + kp + cg + 4, &lds_F[fr][cg + 4]);
        async_wait0();
        __syncthreads();
        const int cl0 = wave * 16;              // wave's 16-col tile in panel
        const int c0  = kp + cl0;
        v8f d;
#pragma unroll
        for (int g = 0; g < 8; ++g) d[g] = lds_A[g + 8 * lhalf][c0 + lrow];
#pragma unroll
        for (int kk = 0; kk < 16; kk += 4) {
          v2f a, bb;
          const float* ap = &lds_coeff[lrow][kk + 2 * lhalf];
          a.x = ap[0]; a.y = ap[1];
          const int fr0 = kk + 2 * lhalf;
          bb.x = lds_F[fr0][cl0 + lrow];        // B[k][c] = F[k][c]
          bb.y = lds_F[fr0 + 1][cl0 + lrow];
          d = __builtin_amdgcn_wmma_f32_16x16x4_f32(
              false, a, false, bb, (short)0, d, false, false);
        }
#pragma unroll
        for (int g = 0; g < 8; ++g) lds_A[g + 8 * lhalf][c0 + lrow] = d[g];
      }
      __syncthreads();
    }

    // ---- Phase 2: sequential Gram-Schmidt inside the 16-row block --------
    for (int i = 0; i < BLK; ++i) {
#pragma unroll
      for (int jj = 0; jj < 2; ++jj) {
        const int j = wave + jj * NWAVES;       // wave-uniform
        if (j < i) {
          float p = 0.f;
          for (int c = lane; c < DIM; c += 32) p += lds_A[j][c] * lds_A[i][c];
#pragma unroll
          for (int off = 16; off > 0; off >>= 1) p += __shfl_xor(p, off, 32);
          if (lane == 0) lds_c1[j] = -p * lds_invsq[base + j];
        }
      }
      __syncthreads();
      float sq = 0.f;
      for (int c = tid; c < DIM; c += TPB) {
        float v = lds_A[i][c];
        for (int j = 0; j < i; ++j) v += lds_c1[j] * lds_A[j][c];
        lds_A[i][c] = v;
        sq += v * v;
      }
      lds_red[tid] = sq;
      __syncthreads();
      for (int s = TPB / 2; s > 0; s >>= 1) {
        if (tid < s) lds_red[tid] += lds_red[tid + s];
        __syncthreads();
      }
      const float total = lds_red[0];
      const float nn    = fmaxf(sqrtf(total), EPS);
      const float scale = lds_norm[base + i] / nn;   // restore original norm
      for (int c = tid; c < DIM; c += TPB) lds_A[i][c] *= scale;
      if (tid == 0)
        lds_invsq[base + i] = 1.f / fmaxf(total * scale * scale, EPS);
      __syncthreads();
    }

    // ---- Phase 3: scatter finalized block to original row positions ------
    {
      float* dstp = out + (size_t)lds_order[base + fr] * DIM;
#pragma unroll 4
      for (int j = 0; j < DIM / 4 / 16; ++j) {      // 32 x b128 per thread
        const int c4 = fsub + j * 16;
        l2g_b128(dstp + 4 * c4, &lds_A[fr][4 * c4]);
      }
      async_wait0();
    }
    __threadfence();   // later blocks re-read these rows from L2
    __syncthreads();
  }
}

extern "C" void kernel_launch(void* const* d_in, const int* in_sizes, int n_in,
                              void* d_out, int out_size, void* d_ws, size_t ws_size,
                              hipStream_t stream) {
  const float* bank    = (const float*)d_in[0];
  const float* fitness = (const float*)d_in[1];
  float* out = (float*)d_out;
  float* ws  = (float*)d_ws;   // needs 768 floats

  row_normsq_kernel<<<N_ROWS, TPB, 0, stream>>>(bank, ws);
  sort_kernel<<<1, N_ROWS, 0, stream>>>(fitness, ws);
  gs_kernel<<<1, TPB, 0, stream>>>(bank, ws, out);
}